// HebbianTraceModule_63316407877695
// MI455X (gfx1250) — compile-verified
//
#include <hip/hip_runtime.h>

typedef __attribute__((ext_vector_type(2))) float v2f;
typedef __attribute__((ext_vector_type(8))) float v8f;
typedef __attribute__((ext_vector_type(4))) unsigned tdm_u4;
typedef __attribute__((ext_vector_type(8))) int      tdm_i8;
typedef __attribute__((ext_vector_type(4))) int      tdm_i4;

#define B_   8
#define H_   8
#define S_   4096
#define D_   64
#define DM_  768
#define ITOT 32752            // B*(S-2)
#define EPS_ 1e-8f

// workspace layout (float offsets)
#define WS_G   0              // H*64*64
#define WS_U   32768          // H*64*64
#define WS_NT  65536          // H*64*64 (new_trace)
#define WS_W2  98304          // 512*768 fused readout matrix

// ---------------------------------------------------------------- TDM 2D tile -> LDS (wave-level DMA)
// Descriptor per CDNA5 ISA ch.8: group0 = {flags, lds_addr, global_addr, type=2},
// group1 = {data_size/pad, tensor dims, tile dims, stride0}; groups 2/3 zero (2D tile).
__device__ __forceinline__ void tdm_load_2d(unsigned lds_byte, const float* gsrc,
                                            unsigned td0, unsigned td1,
                                            unsigned tile0, unsigned tile1,
                                            unsigned stride0,
                                            unsigned pad_int_code, unsigned pad_amt_code) {
    unsigned long long ga = (unsigned long long)(size_t)gsrc;
    tdm_u4 g0;
    g0.x = 1u;                                                   // count=1 (valid user D#)
    g0.y = lds_byte;                                             // lds_addr
    g0.z = (unsigned)ga;                                         // global_addr[31:0]
    g0.w = ((unsigned)(ga >> 32) & 0x1ffffffu) | (2u << 30);     // global_addr[56:32] | type=2
    tdm_i8 g1;
    g1[0] = (int)((2u << 16)                                     // data_size = 4B
                | (1u << 20)                                     // pad_enable
                | (pad_int_code << 22)                           // pad_interval
                | (pad_amt_code << 25));                         // pad_amount
    g1[1] = (int)((td0 & 0xffffu) << 16);                        // tensor_dim0[15:0]
    g1[2] = (int)(((td0 >> 16) & 0xffffu) | ((td1 & 0xffffu) << 16));
    g1[3] = (int)(((td1 >> 16) & 0xffffu) | ((tile0 & 0xffffu) << 16));
    g1[4] = (int)(tile1 & 0xffffu);                              // tile_dim1 (tile_dim2 = 0)
    g1[5] = (int)stride0;                                        // tensor_dim0_stride[31:0]
    g1[6] = 0;
    g1[7] = 0;
    tdm_i4 gz4 = {0, 0, 0, 0};
    tdm_i8 gz8 = {0, 0, 0, 0, 0, 0, 0, 0};
    __builtin_amdgcn_tensor_load_to_lds(g0, g1, gz4, gz4, gz8, 0);
}

// ---------------------------------------------------------------- zero G,U
__global__ void k_zero(float* ws) {
    ws[blockIdx.x * blockDim.x + threadIdx.x] = 0.0f;   // 65536 threads
}

// ---------------------------------------------------------------- Gram + update reduction
// G[h] = sum_rows (q/||q||)(q/||q||)^T ; U[h] = sum_rows q v^T   (row = (b,i), i<S-2, v = V[b,h,i+2])
#define GR_ROWS   64
#define GR_STRIDE 80          // (16r+c)%64 bank pattern: lane halves (rows +2) hit disjoint banks
#define GR_STAGES 16
#define GR_CHUNK  (GR_ROWS * GR_STAGES)   // 1024 rows per block
#define GR_NCH    32                      // ceil(32752/1024)

__global__ __launch_bounds__(512) void k_gram(const float* __restrict__ Q,
                                              const float* __restrict__ V,
                                              float* __restrict__ ws) {
    __shared__ float Qs[GR_ROWS * GR_STRIDE];
    __shared__ float Vs[GR_ROWS * GR_STRIDE];
    __shared__ float sinv[GR_ROWS];

    const int h    = blockIdx.x;
    const int tid  = threadIdx.x;
    const int lane = tid & 31;
    const int wave = tid >> 5;
    const int lo   = lane & 15;
    const int hi   = lane >> 4;
    const int p0   = (wave >> 2) * 16;    // 16 waves cover the 4x4 grid of 16x16 tiles
    const int q0   = (wave & 3) * 16;

    v8f accg = {0.f,0.f,0.f,0.f,0.f,0.f,0.f,0.f};
    v8f accu = {0.f,0.f,0.f,0.f,0.f,0.f,0.f,0.f};

    for (int st = 0; st < GR_STAGES; ++st) {
        const int row_base = blockIdx.y * GR_CHUNK + st * GR_ROWS;
        if (tid < GR_ROWS) sinv[tid] = 0.0f;
        __syncthreads();
        // stage 64x64 of Q_store and V_store (shift +2); accumulate row sum-of-squares
        #pragma unroll
        for (int k = 0; k < 2; ++k) {
            int id  = tid + k * 512;          // 1024 float4 total
            int row = id >> 4;
            int seg = id & 15;
            int rg  = row_base + row;
            float4 q4 = make_float4(0.f, 0.f, 0.f, 0.f);
            float4 v4 = make_float4(0.f, 0.f, 0.f, 0.f);
            if (rg < ITOT) {
                int b = rg / (S_ - 2);
                int i = rg - b * (S_ - 2);
                size_t base = (((size_t)b * H_ + h) * S_);
                q4 = *(const float4*)(Q + ((base + i)     * D_ + seg * 4));
                v4 = *(const float4*)(V + ((base + i + 2) * D_ + seg * 4));
            }
            *(float4*)&Qs[row * GR_STRIDE + seg * 4] = q4;
            *(float4*)&Vs[row * GR_STRIDE + seg * 4] = v4;
            atomicAdd(&sinv[row], q4.x*q4.x + q4.y*q4.y + q4.z*q4.z + q4.w*q4.w);
        }
        __syncthreads();
        if (tid < GR_ROWS) sinv[tid] = 1.0f / fmaxf(sqrtf(sinv[tid]), EPS_);
        __syncthreads();
        // K-loop over the 64 staged rows, 4 at a time (V_WMMA_F32_16X16X4_F32)
        for (int kk = 0; kk < GR_ROWS; kk += 4) {
            int r0 = kk + 2 * hi;             // A 16x4 layout: lanes 16-31 carry K=2,3
            int r1 = r0 + 1;
            float s0 = sinv[r0], s1 = sinv[r1];
            float qp0 = Qs[r0 * GR_STRIDE + p0 + lo];
            float qp1 = Qs[r1 * GR_STRIDE + p0 + lo];
            float qq0 = Qs[r0 * GR_STRIDE + q0 + lo];
            float qq1 = Qs[r1 * GR_STRIDE + q0 + lo];
            float vq0 = Vs[r0 * GR_STRIDE + q0 + lo];
            float vq1 = Vs[r1 * GR_STRIDE + q0 + lo];
            v2f au; au.x = qp0;      au.y = qp1;
            v2f bu; bu.x = vq0;      bu.y = vq1;
            accu = __builtin_amdgcn_wmma_f32_16x16x4_f32(false, au, false, bu,
                                                         (short)0, accu, false, false);
            v2f ag; ag.x = qp0 * s0; ag.y = qp1 * s1;
            v2f bg; bg.x = qq0 * s0; bg.y = qq1 * s1;
            accg = __builtin_amdgcn_wmma_f32_16x16x4_f32(false, ag, false, bg,
                                                         (short)0, accg, false, false);
        }
        __syncthreads();
    }
    // reduce partial 16x16 tiles into global accumulators
    float* G = ws + WS_G + h * 4096;
    float* U = ws + WS_U + h * 4096;
    #pragma unroll
    for (int j = 0; j < 8; ++j) {            // C layout: VGPR j -> row j (+8 for lanes 16-31)
        int pr = p0 + j + 8 * hi;
        int qc = q0 + lo;
        atomicAdd(&G[pr * 64 + qc], accg[j]);
        atomicAdd(&U[pr * 64 + qc], accu[j]);
    }
}

// ---------------------------------------------------------------- new_trace = .99*(T - G@T/denom) + .1*U/denom
__global__ __launch_bounds__(256) void k_trace(const float* __restrict__ trace,
                                               float* __restrict__ ws) {
    __shared__ float Gs[64 * 68];
    __shared__ float Ts[64 * 68];
    const int h   = blockIdx.x;
    const int tid = threadIdx.x;
    const float* G  = ws + WS_G  + h * 4096;
    const float* U  = ws + WS_U  + h * 4096;
    const float* T  = trace      + h * 4096;
    float*       NT = ws + WS_NT + h * 4096;
    #pragma unroll
    for (int k = 0; k < 4; ++k) {
        int id = tid + k * 256;              // 1024 float4
        int row = id >> 4, seg = id & 15;
        *(float4*)&Gs[row * 68 + seg * 4] = *(const float4*)&G[row * 64 + seg * 4];
        *(float4*)&Ts[row * 68 + seg * 4] = *(const float4*)&T[row * 64 + seg * 4];
    }
    __syncthreads();
    const float invd = 1.0f / (float)ITOT;
    const int p  = tid >> 2;
    const int qb = (tid & 3) * 16;
    float acc[16];
    #pragma unroll
    for (int j = 0; j < 16; ++j) acc[j] = 0.0f;
    for (int k = 0; k < 64; ++k) {
        float g = Gs[p * 68 + k];
        #pragma unroll
        for (int j = 0; j < 16; ++j)
            acc[j] = fmaf(g, Ts[k * 68 + qb + j], acc[j]);
    }
    #pragma unroll
    for (int j = 0; j < 16; ++j) {
        int q = qb + j;
        NT[p * 64 + q] = 0.99f * (Ts[p * 68 + q] - acc[j] * invd)
                       + 0.10f * U[p * 64 + q] * invd;
    }
}

// ---------------------------------------------------------------- W2[k][m] = sum_q NT[k][q] * W_out[m][(k&~63)+q]
__global__ void k_w2(const float* __restrict__ Wout, float* __restrict__ ws) {
    int g = blockIdx.x * blockDim.x + threadIdx.x;   // 512*768 threads
    int k = g / DM_;
    int m = g - k * DM_;
    const float* nt = ws + WS_NT + k * 64;
    const float* wr = Wout + (size_t)m * 512 + (k & ~63);
    float acc = 0.0f;
    #pragma unroll 8
    for (int q = 0; q < 64; ++q) acc = fmaf(nt[q], wr[q], acc);
    ws[WS_W2 + (size_t)k * DM_ + m] = acc;
}

// ---------------------------------------------------------------- out(32768x768) = Qaddr_flat(32768x512) @ W2(512x768)
// TDM double-buffered: stage i+1 DMA overlaps stage i WMMA.
#define MT  128
#define NTL 128
#define KT  32
#define A_STR 36      // 32 + pad 4  -> (36r+c)%64 bank pattern  (TDM pad codes 4 / 3)
#define B_STR 144     // 128 + pad 16 -> (16r+c)%64 bank pattern (TDM pad codes 6 / 15)
#define A_BUF (MT * A_STR)
#define B_BUF (KT * B_STR)

__global__ __launch_bounds__(512) void k_out(const float* __restrict__ Q,
                                             const float* __restrict__ ws,
                                             float* __restrict__ out) {
    __shared__ float As[2 * A_BUF];
    __shared__ float Bs[2 * B_BUF];
    const float* W2 = ws + WS_W2;

    const int tid  = threadIdx.x;
    const int lane = tid & 31, wave = tid >> 5;
    const int lo   = lane & 15, hi = lane >> 4;
    const int bs0  = blockIdx.x * MT;
    const int m0   = blockIdx.y * NTL;
    const int b    = bs0 >> 12;
    const int s0   = bs0 & 4095;
    const bool bnd = (s0 == 0);               // first row is the Q_addr zero row
    const int rb   = (wave >> 2) * 32;
    const int cb   = (wave & 3) * 32;

    const unsigned lds_a = (unsigned)(size_t)(void*)As;
    const unsigned lds_b = (unsigned)(size_t)(void*)Bs;

    // boundary blocks: keep LDS row 0 of both A buffers at zero; TDM writes rows 1..127
    if (bnd && tid < A_STR) { As[tid] = 0.0f; As[A_BUF + tid] = 0.0f; }

    auto issue = [&](int kt, int buf) {
        const int hh = kt >> 6, pp = kt & 63;
        const float* qa = Q + ((((size_t)b * H_ + hh) * S_ + (bnd ? 0 : (s0 - 1))) * D_ + pp);
        unsigned la = lds_a + (unsigned)(buf * A_BUF + (bnd ? A_STR : 0)) * 4u;
        unsigned rows = bnd ? 127u : 128u;
        tdm_load_2d(la, qa, 32u, rows, 32u, rows, 64u, 4u, 3u);          // A: 128x32, stride 64
        unsigned lb = lds_b + (unsigned)(buf * B_BUF) * 4u;
        tdm_load_2d(lb, W2 + (size_t)kt * DM_ + m0,
                    128u, 32u, 128u, 32u, (unsigned)DM_, 6u, 15u);       // B: 32x128, stride 768
    };

    if (wave == 0) issue(0, 0);

    v8f acc00 = {0.f,0.f,0.f,0.f,0.f,0.f,0.f,0.f};
    v8f acc01 = acc00, acc10 = acc00, acc11 = acc00;

    for (int it = 0; it < 16; ++it) {
        const int cur = it & 1;
        if (wave == 0) {
            if (it + 1 < 16) {
                issue((it + 1) * KT, 1 - cur);           // prefetch next stage
                __builtin_amdgcn_s_wait_tensorcnt(2);    // retire current stage (in-order)
            } else {
                __builtin_amdgcn_s_wait_tensorcnt(0);    // last stage: drain
            }
        }
        __syncthreads();
        const float* Ab = As + cur * A_BUF;
        const float* Bb = Bs + cur * B_BUF;
        #pragma unroll
        for (int kk = 0; kk < KT; kk += 4) {
            int kc = kk + 2 * hi;
            v2f a0 = *(const v2f*)&Ab[(rb + lo)      * A_STR + kc];
            v2f a1 = *(const v2f*)&Ab[(rb + 16 + lo) * A_STR + kc];
            v2f b0; b0.x = Bb[kc * B_STR + cb + lo];      b0.y = Bb[(kc + 1) * B_STR + cb + lo];
            v2f b1; b1.x = Bb[kc * B_STR + cb + 16 + lo]; b1.y = Bb[(kc + 1) * B_STR + cb + 16 + lo];
            acc00 = __builtin_amdgcn_wmma_f32_16x16x4_f32(false, a0, false, b0, (short)0, acc00, false, false);
            acc01 = __builtin_amdgcn_wmma_f32_16x16x4_f32(false, a0, false, b1, (short)0, acc01, false, false);
            acc10 = __builtin_amdgcn_wmma_f32_16x16x4_f32(false, a1, false, b0, (short)0, acc10, false, false);
            acc11 = __builtin_amdgcn_wmma_f32_16x16x4_f32(false, a1, false, b1, (short)0, acc11, false, false);
        }
        __syncthreads();
    }
    #pragma unroll
    for (int j = 0; j < 8; ++j) {
        int r0 = bs0 + rb + j + 8 * hi;
        int r1 = r0 + 16;
        int c0 = m0 + cb + lo;
        int c1 = c0 + 16;
        out[(size_t)r0 * DM_ + c0] = acc00[j];
        out[(size_t)r0 * DM_ + c1] = acc01[j];
        out[(size_t)r1 * DM_ + c0] = acc10[j];
        out[(size_t)r1 * DM_ + c1] = acc11[j];
    }
}

// ---------------------------------------------------------------- launch
extern "C" void kernel_launch(void* const* d_in, const int* in_sizes, int n_in,
                              void* d_out, int out_size, void* d_ws, size_t ws_size,
                              hipStream_t stream) {
    const float* Q     = (const float*)d_in[0];
    const float* V     = (const float*)d_in[1];
    const float* trace = (const float*)d_in[2];
    const float* Wout  = (const float*)d_in[3];
    float* out = (float*)d_out;
    float* ws  = (float*)d_ws;

    k_zero<<<256, 256, 0, stream>>>(ws);                     // zero G,U accumulators
    k_gram<<<dim3(H_, GR_NCH), 512, 0, stream>>>(Q, V, ws);  // WMMA Gram + update reduction
    k_trace<<<H_, 256, 0, stream>>>(trace, ws);              // erase/decay -> new_trace
    k_w2<<<(512 * DM_) / 256, 256, 0, stream>>>(Wout, ws);   // fuse new_trace with W_out
    k_out<<<dim3(32768 / MT, DM_ / NTL), 512, 0, stream>>>(Q, ws, out); // TDM + WMMA main GEMM
}